// MambaEncoder_75402445848781
// MI455X (gfx1250) — compile-verified
//
#include <hip/hip_runtime.h>
#include <math.h>

typedef float v2f __attribute__((ext_vector_type(2)));
typedef float v8f __attribute__((ext_vector_type(8)));

namespace {

constexpr int L_LAYERS = 2;
constexpr int DIN  = 64;
constexpr int DM   = 256;
constexpr int DI   = 512;     // 2*DM
constexpr int NST  = 16;      // state dim
constexpr int KCONV= 4;
constexpr int RLOW = 16;      // DM/16
constexpr int DOUT = 128;
constexpr int BB   = 8;
constexpr int TT   = 2048;
constexpr int MROWS= BB * TT; // 16384

__device__ __forceinline__ float silu_f(float v) {
    return v / (1.0f + expf(-v));
}
__device__ __forceinline__ float softplus_f(float v) {
    if (v > 20.0f) return v;
    return log1pf(expf(v));
}

// ---------------------------------------------------------------------------
// Tiled fp32 GEMM on V_WMMA_F32_16X16X4_F32.
// Wave tile: 32 rows x (NT*16) cols. EP: 0 = none, 1 = +bias, 2 = +bias,softplus
// A: MxK row-major (lda), B: KxN row-major (ldb), C: MxN row-major (ldc).
// M must be a multiple of 32, K a multiple of 4, N a multiple of NT*16.
// ---------------------------------------------------------------------------
template <int NT, int EP>
__global__ void gemm_wmma_f32(const float* __restrict__ A,
                              const float* __restrict__ B,
                              float* __restrict__ C,
                              const float* __restrict__ bias,
                              int M, int K, int N,
                              int lda, int ldb, int ldc,
                              int tilesN)
{
    const int wave = (int)((blockIdx.x * blockDim.x + threadIdx.x) >> 5);
    const int lane = (int)(threadIdx.x & 31u);
    const int tn   = wave % tilesN;
    const int tm   = wave / tilesN;
    if (tm * 32 >= M) return;  // uniform per wave: EXEC stays all-ones

    const int row0  = tm * 32;
    const int col0  = tn * (NT * 16);
    const int mhalf = lane >> 4;    // 0 or 1
    const int mlo   = lane & 15;

    v8f acc[2][NT] = {};

    for (int k = 0; k < K; k += 4) {
        const int kk = k + 2 * mhalf;   // this lane's K pair: kk, kk+1

        // A fragments: lane holds A[row0(+16)+mlo][kk..kk+1] (8B aligned: lda,kk even)
        v2f a0 = *(const v2f*)(A + (size_t)(row0 + mlo)      * lda + kk);
        v2f a1 = *(const v2f*)(A + (size_t)(row0 + 16 + mlo) * lda + kk);

        // B fragments: lane holds B[kk..kk+1][col0 + j*16 + mlo]
        v2f bf[NT];
#pragma unroll
        for (int j = 0; j < NT; ++j) {
            const int col = col0 + j * 16 + mlo;
            bf[j][0] = B[(size_t)kk       * ldb + col];
            bf[j][1] = B[(size_t)(kk + 1) * ldb + col];
        }

        if (k + 4 < K)  // stream B through: CDNA5 global_prefetch_b8
            __builtin_prefetch(B + (size_t)(kk + 4) * ldb + col0 + mlo, 0, 1);

#pragma unroll
        for (int j = 0; j < NT; ++j) {
            acc[0][j] = __builtin_amdgcn_wmma_f32_16x16x4_f32(
                false, a0, false, bf[j], (short)0, acc[0][j], false, false);
            acc[1][j] = __builtin_amdgcn_wmma_f32_16x16x4_f32(
                false, a1, false, bf[j], (short)0, acc[1][j], false, false);
        }
    }

#pragma unroll
    for (int mi = 0; mi < 2; ++mi) {
#pragma unroll
        for (int j = 0; j < NT; ++j) {
            const int col = col0 + j * 16 + mlo;
            float bv = 0.0f;
            if (EP >= 1) bv = bias[col];
#pragma unroll
            for (int r = 0; r < 8; ++r) {
                const int row = row0 + mi * 16 + r + 8 * mhalf;
                float v = acc[mi][j][r] + bv;
                if (EP == 2) v = softplus_f(v);
                C[(size_t)row * ldc + col] = v;
            }
        }
    }
}

// ---------------------------------------------------------------------------
// Depthwise causal conv (K=4) + bias + SiLU.  xi lives in XZ cols [0,DI).
// ---------------------------------------------------------------------------
__global__ void conv_silu_kernel(const float* __restrict__ xz,
                                 const float* __restrict__ cw,   // (DI,4)
                                 const float* __restrict__ cb,   // (DI,)
                                 float* __restrict__ xc)
{
    const size_t idx = (size_t)blockIdx.x * blockDim.x + threadIdx.x;
    if (idx >= (size_t)MROWS * DI) return;
    const int d   = (int)(idx & (DI - 1));
    const int row = (int)(idx >> 9);        // b*T + t
    const int t   = row & (TT - 1);

    float s = cb[d];
#pragma unroll
    for (int k = 0; k < KCONV; ++k) {
        const int tt = t + k - (KCONV - 1);
        if (tt >= 0)
            s += xz[(size_t)(row + k - (KCONV - 1)) * (2 * DI) + d] * cw[d * KCONV + k];
    }
    xc[idx] = silu_f(s);
}

// ---------------------------------------------------------------------------
// Selective scan: one thread per (b, d); B_t / C_t staged via LDS in chunks.
// ---------------------------------------------------------------------------
__global__ void scan_kernel(const float* __restrict__ dbc,     // (B*T, 48)
                            const float* __restrict__ dt,      // (B*T, DI)
                            const float* __restrict__ xc,      // (B*T, DI)
                            const float* __restrict__ a_log,   // (DI, NST)
                            float* __restrict__ ys)            // (B*T, DI)
{
    const int b = blockIdx.x;
    const int d = threadIdx.x;      // 512 threads

    __shared__ float sBC[16][2 * NST];  // [chunk step][B(0..15) C(16..31)]

    float a_n[NST];
#pragma unroll
    for (int n = 0; n < NST; ++n) a_n[n] = -expf(a_log[d * NST + n]);

    float hc[NST];
#pragma unroll
    for (int n = 0; n < NST; ++n) hc[n] = 0.0f;

    for (int t0 = 0; t0 < TT; t0 += 16) {
        __syncthreads();
        {   // 512 threads fill 16x32 values: B/C columns 16..47 of dbc
            const int i = d >> 5;
            const int j = d & 31;
            sBC[i][j] = dbc[(size_t)(b * TT + t0 + i) * 48 + RLOW + j];
        }
        __syncthreads();

        for (int i = 0; i < 16; ++i) {
            const size_t off = (size_t)(b * TT + t0 + i) * DI + d;
            const float dtv  = dt[off];
            const float xv   = xc[off];
            const float coef = dtv * xv;
            float y = 0.0f;
#pragma unroll
            for (int n = 0; n < NST; ++n) {
                const float dA = __expf(dtv * a_n[n]);
                hc[n] = dA * hc[n] + coef * sBC[i][n];
                y += hc[n] * sBC[i][NST + n];
            }
            ys[off] = y;
        }
    }
}

// ---------------------------------------------------------------------------
// y = (ys + Dp*xc) * silu(z), in place on ys.  z lives in XZ cols [DI, 2*DI).
// ---------------------------------------------------------------------------
__global__ void gate_kernel(const float* __restrict__ xz,
                            const float* __restrict__ xc,
                            const float* __restrict__ dp,
                            float* __restrict__ ys)
{
    const size_t idx = (size_t)blockIdx.x * blockDim.x + threadIdx.x;
    if (idx >= (size_t)MROWS * DI) return;
    const int d   = (int)(idx & (DI - 1));
    const int row = (int)(idx >> 9);
    const float zv = xz[(size_t)row * (2 * DI) + DI + d];
    ys[idx] = (ys[idx] + dp[d] * xc[idx]) * silu_f(zv);
}

// ---------------------------------------------------------------------------
// Mean over T:  hmean[b, m] = (1/T) * sum_t h[b, t, m]
// ---------------------------------------------------------------------------
__global__ void mean_kernel(const float* __restrict__ h, float* __restrict__ hmean)
{
    const int b = blockIdx.x;
    const int m = threadIdx.x;  // 256
    float s = 0.0f;
    for (int t = 0; t < TT; ++t)
        s += h[(size_t)(b * TT + t) * DM + m];
    hmean[b * DM + m] = s * (1.0f / (float)TT);
}

// ---------------------------------------------------------------------------
// out[b, j] = hmean[b, :] . W_out_proj[:, j] + b_out_proj[j]
// ---------------------------------------------------------------------------
__global__ void outproj_kernel(const float* __restrict__ hmean,
                               const float* __restrict__ w,     // (DM, DOUT)
                               const float* __restrict__ bias,  // (DOUT,)
                               float* __restrict__ out)
{
    const int idx = (int)(blockIdx.x * blockDim.x + threadIdx.x);
    if (idx >= BB * DOUT) return;
    const int b = idx >> 7;
    const int j = idx & (DOUT - 1);
    float s = bias[j];
    for (int m = 0; m < DM; ++m)
        s += hmean[b * DM + m] * w[m * DOUT + j];
    out[idx] = s;
}

template <int NT, int EP>
static void run_gemm(const float* A, const float* B, float* C, const float* bias,
                     int M, int K, int N, int lda, int ldb, int ldc, hipStream_t s)
{
    const int tilesN = N / (NT * 16);
    const long waves = (long)(M / 32) * tilesN;
    const long threads = waves * 32;
    const int block = 256;
    const int grid = (int)((threads + block - 1) / block);
    hipLaunchKernelGGL((gemm_wmma_f32<NT, EP>), dim3(grid), dim3(block), 0, s,
                       A, B, C, bias, M, K, N, lda, ldb, ldc, tilesN);
}

} // namespace

extern "C" void kernel_launch(void* const* d_in, const int* in_sizes, int n_in,
                              void* d_out, int out_size, void* d_ws, size_t ws_size,
                              hipStream_t stream)
{
    const float* x    = (const float*)d_in[0];   // (B,T,DIN)
    const float* Wip  = (const float*)d_in[1];   // (DIN,DM)
    const float* bip  = (const float*)d_in[2];   // (DM,)
    const float* Win  = (const float*)d_in[3];   // (L,DM,2*DI)
    const float* bin_ = (const float*)d_in[4];   // (L,2*DI)
    const float* cw   = (const float*)d_in[5];   // (L,DI,K)
    const float* cb   = (const float*)d_in[6];   // (L,DI)
    const float* Wx   = (const float*)d_in[7];   // (L,DI,R+2N)=(L,512,48)
    const float* Wdt  = (const float*)d_in[8];   // (L,R,DI)
    const float* bdt  = (const float*)d_in[9];   // (L,DI)
    const float* Alog = (const float*)d_in[10];  // (L,DI,NST)
    const float* Dp   = (const float*)d_in[11];  // (L,DI)
    const float* Wout = (const float*)d_in[12];  // (L,DI,DM)
    const float* Wop  = (const float*)d_in[13];  // (DM,DOUT)
    const float* bop  = (const float*)d_in[14];  // (DOUT,)
    float* out        = (float*)d_out;

    // Workspace layout (floats)
    float* ws = (float*)d_ws;
    const size_t SZ_H   = (size_t)MROWS * DM;        // 4,194,304
    const size_t SZ_XZ  = (size_t)MROWS * 2 * DI;    // 16,777,216
    const size_t SZ_DI  = (size_t)MROWS * DI;        // 8,388,608
    const size_t SZ_DBC = (size_t)MROWS * 48;        //   786,432
    float* H0    = ws;
    float* H1    = H0  + SZ_H;
    float* XZ    = H1  + SZ_H;
    float* XC    = XZ  + SZ_XZ;
    float* DT    = XC  + SZ_DI;
    float* YS    = DT  + SZ_DI;
    float* DBC   = YS  + SZ_DI;
    float* HMEAN = DBC + SZ_DBC;

    const int blk = 256;
    const int gridEl = (int)(((size_t)MROWS * DI + blk - 1) / blk);

    // h = x @ W_in_proj + b
    run_gemm<4, 1>(x, Wip, H0, bip, MROWS, DIN, DM, DIN, DM, DM, stream);

    for (int l = 0; l < L_LAYERS; ++l) {
        const float* hin = (l & 1) ? H1 : H0;
        float* hout      = (l & 1) ? H0 : H1;

        // xz = h @ Win[l] + bin[l]
        run_gemm<4, 1>(hin, Win + (size_t)l * DM * 2 * DI, XZ, bin_ + (size_t)l * 2 * DI,
                       MROWS, DM, 2 * DI, DM, 2 * DI, 2 * DI, stream);

        // xc = silu(depthwise_conv(xi) + conv_b)
        hipLaunchKernelGGL(conv_silu_kernel, dim3(gridEl), dim3(blk), 0, stream,
                           XZ, cw + (size_t)l * DI * KCONV, cb + (size_t)l * DI, XC);

        // dbc = xc @ Wx[l]                          (N = 48 -> NT = 3)
        run_gemm<3, 0>(XC, Wx + (size_t)l * DI * 48, DBC, nullptr,
                       MROWS, DI, 48, DI, 48, 48, stream);

        // dt = softplus(dbc[:, :R] @ Wdt[l] + bdt[l])   (K = 16, lda = 48)
        run_gemm<4, 2>(DBC, Wdt + (size_t)l * RLOW * DI, DT, bdt + (size_t)l * DI,
                       MROWS, RLOW, DI, 48, DI, DI, stream);

        // selective state-space scan -> ys
        hipLaunchKernelGGL(scan_kernel, dim3(BB), dim3(DI), 0, stream,
                           DBC, DT, XC, Alog + (size_t)l * DI * NST, YS);

        // ys = (ys + Dp*xc) * silu(z)
        hipLaunchKernelGGL(gate_kernel, dim3(gridEl), dim3(blk), 0, stream,
                           XZ, XC, Dp + (size_t)l * DI, YS);

        // h_next = ys @ Wout[l]
        run_gemm<4, 0>(YS, Wout + (size_t)l * DI * DM, hout, nullptr,
                       MROWS, DI, DM, DI, DM, DM, stream);
    }

    const float* hfinal = (L_LAYERS & 1) ? H1 : H0;   // L=2 -> H0
    hipLaunchKernelGGL(mean_kernel, dim3(BB), dim3(DM), 0, stream, hfinal, HMEAN);
    hipLaunchKernelGGL(outproj_kernel, dim3((BB * DOUT + blk - 1) / blk), dim3(blk), 0, stream,
                       HMEAN, Wop, bop, out);
}